// ConformerEncoder_89678917141137
// MI455X (gfx1250) — compile-verified
//
#include <hip/hip_runtime.h>
#include <hip/hip_bf16.h>
#include <cstddef>

#define EPSV 1e-5f
#define NHEADS 4
#define SEQ 256
#define DMODEL 144
#define HDIM 36
#define NTOK 2048      // B * SEQ
#define NLAYER 8
#define FFDIM 576
#define NBATCH 8

typedef float v2f __attribute__((ext_vector_type(2)));
typedef float v8f __attribute__((ext_vector_type(8)));

__device__ __forceinline__ float sigmoid_f(float x) { return 1.0f / (1.0f + __expf(-x)); }

// ---------------------------------------------------------------------------
// WMMA fp32 GEMM:  out[M,N] = epilogue( A[M,K] @ W[N,K]^T + bias )
// epilogue: v = (acc + bias) * post_scale; act(1=swish); if addto: v = addto + add_scale*v
// One wave computes a 64x16 tile: 4 M-sub-tiles share one B fragment per K-step.
// Software-pipelined: fragments for k-step i+1 are fetched into fresh locals
// before the 4 WMMAs of step i are issued (keeps loads in flight, partial waits).
// A fragment: lane m = lane&15, k = (lane>>4)*2 + {0,1}  (float2 load)
// B fragment: lane n = lane&15, same k mapping           (float2 load)
// D fragment: lane n = lane&15, VGPR r -> m = 8*(lane>>4) + r
// M must be a multiple of 64, N of 16, K of 4 (and K >= 4).
// ---------------------------------------------------------------------------
__global__ void gemm_wmma(const float* __restrict__ A, const float* __restrict__ W,
                          const float* __restrict__ bias, float* __restrict__ out,
                          const float* __restrict__ addto,
                          int M, int N, int K, float post_scale, int act, float add_scale) {
  int lane = threadIdx.x & 31;
  int mBase = blockIdx.x * 64;
  int nBase = blockIdx.y * 16;
  int r = lane & 15;
  int g = lane >> 4;
  const float* aRow = A + (size_t)(mBase + r) * K + 2 * g;
  const float* bRow = W + (size_t)(nBase + r) * K + 2 * g;
  size_t aS = (size_t)16 * K;
  v8f acc0 = {}, acc1 = {}, acc2 = {}, acc3 = {};

  v2f bC = *(const v2f*)(bRow);
  v2f aC0 = *(const v2f*)(aRow);
  v2f aC1 = *(const v2f*)(aRow + aS);
  v2f aC2 = *(const v2f*)(aRow + 2 * aS);
  v2f aC3 = *(const v2f*)(aRow + 3 * aS);

  int k0 = 0;
  for (; k0 + 4 < K; k0 += 4) {
    int kn = k0 + 4;
    v2f bN = *(const v2f*)(bRow + kn);
    v2f aN0 = *(const v2f*)(aRow + kn);
    v2f aN1 = *(const v2f*)(aRow + aS + kn);
    v2f aN2 = *(const v2f*)(aRow + 2 * aS + kn);
    v2f aN3 = *(const v2f*)(aRow + 3 * aS + kn);
    acc0 = __builtin_amdgcn_wmma_f32_16x16x4_f32(false, aC0, false, bC, (short)0, acc0, false, false);
    acc1 = __builtin_amdgcn_wmma_f32_16x16x4_f32(false, aC1, false, bC, (short)0, acc1, false, false);
    acc2 = __builtin_amdgcn_wmma_f32_16x16x4_f32(false, aC2, false, bC, (short)0, acc2, false, false);
    acc3 = __builtin_amdgcn_wmma_f32_16x16x4_f32(false, aC3, false, bC, (short)0, acc3, false, false);
    bC = bN; aC0 = aN0; aC1 = aN1; aC2 = aN2; aC3 = aN3;
  }
  acc0 = __builtin_amdgcn_wmma_f32_16x16x4_f32(false, aC0, false, bC, (short)0, acc0, false, false);
  acc1 = __builtin_amdgcn_wmma_f32_16x16x4_f32(false, aC1, false, bC, (short)0, acc1, false, false);
  acc2 = __builtin_amdgcn_wmma_f32_16x16x4_f32(false, aC2, false, bC, (short)0, acc2, false, false);
  acc3 = __builtin_amdgcn_wmma_f32_16x16x4_f32(false, aC3, false, bC, (short)0, acc3, false, false);

  int n = nBase + r;
  float bv = bias ? bias[n] : 0.0f;
  v8f accs[4] = {acc0, acc1, acc2, acc3};
#pragma unroll
  for (int t = 0; t < 4; ++t) {
#pragma unroll
    for (int i = 0; i < 8; ++i) {
      int m = mBase + 16 * t + 8 * g + i;
      float v = (accs[t][i] + bv) * post_scale;
      if (act == 1) v = v * sigmoid_f(v);
      if (addto) v = addto[(size_t)m * N + n] + add_scale * v;
      out[(size_t)m * N + n] = v;
    }
  }
}

// ---------------------------------------------------------------------------
// Implicit-GEMM conv2d (single batch, NCHW in / CHW out), cross-correlation,
// zero padding. M = H*W output pixels (mult of 64), N = Cout (mult of 16),
// K = Cin*Ks*Ks (zero-padded to mult of 4). 64x16 tile per wave; per-k decode
// shared across the 4 M-sub-tiles; software-pipelined gathers.
// ---------------------------------------------------------------------------
__global__ void conv_wmma(const float* __restrict__ in, const float* __restrict__ wgt,
                          const float* __restrict__ bias, float* __restrict__ out,
                          int Cin, int H, int W, int Cout, int Ks, int pad) {
  int lane = threadIdx.x & 31;
  int mBase = blockIdx.x * 64;
  int nBase = blockIdx.y * 16;
  int r = lane & 15;
  int g = lane >> 4;
  int M = H * W;
  int oh[4], ow[4];
#pragma unroll
  for (int t = 0; t < 4; ++t) {
    int m = mBase + 16 * t + r;
    oh[t] = m / W;
    ow[t] = m - oh[t] * W;
  }
  int KK = Ks * Ks;
  int Ktot = Cin * KK;
  const float* wRow = wgt + (size_t)(nBase + r) * Ktot;

  auto fetch = [&](int k0, v2f& bo, v2f* ao) {
#pragma unroll
    for (int vi = 0; vi < 2; ++vi) {
      int kk = k0 + 2 * g + vi;
      float bvw = 0.0f;
      int ci = 0, kh = 0, kw = 0;
      bool kok = kk < Ktot;
      if (kok) {
        ci = kk / KK;
        int rem = kk - ci * KK;
        kh = rem / Ks;
        kw = rem - kh * Ks;
        bvw = wRow[kk];
      }
      bo[vi] = bvw;
#pragma unroll
      for (int t = 0; t < 4; ++t) {
        float av = 0.0f;
        if (kok) {
          int ih = oh[t] - pad + kh;
          int iw = ow[t] - pad + kw;
          if (ih >= 0 && ih < H && iw >= 0 && iw < W)
            av = in[(size_t)ci * M + (size_t)ih * W + iw];
        }
        ao[t][vi] = av;
      }
    }
  };

  v8f acc0 = {}, acc1 = {}, acc2 = {}, acc3 = {};
  v2f bC, aC[4];
  fetch(0, bC, aC);
  int k0 = 0;
  for (; k0 + 4 < Ktot; k0 += 4) {
    v2f bN, aN[4];
    fetch(k0 + 4, bN, aN);
    acc0 = __builtin_amdgcn_wmma_f32_16x16x4_f32(false, aC[0], false, bC, (short)0, acc0, false, false);
    acc1 = __builtin_amdgcn_wmma_f32_16x16x4_f32(false, aC[1], false, bC, (short)0, acc1, false, false);
    acc2 = __builtin_amdgcn_wmma_f32_16x16x4_f32(false, aC[2], false, bC, (short)0, acc2, false, false);
    acc3 = __builtin_amdgcn_wmma_f32_16x16x4_f32(false, aC[3], false, bC, (short)0, acc3, false, false);
    bC = bN; aC[0] = aN[0]; aC[1] = aN[1]; aC[2] = aN[2]; aC[3] = aN[3];
  }
  acc0 = __builtin_amdgcn_wmma_f32_16x16x4_f32(false, aC[0], false, bC, (short)0, acc0, false, false);
  acc1 = __builtin_amdgcn_wmma_f32_16x16x4_f32(false, aC[1], false, bC, (short)0, acc1, false, false);
  acc2 = __builtin_amdgcn_wmma_f32_16x16x4_f32(false, aC[2], false, bC, (short)0, acc2, false, false);
  acc3 = __builtin_amdgcn_wmma_f32_16x16x4_f32(false, aC[3], false, bC, (short)0, acc3, false, false);

  int n = nBase + r;
  float bb = bias[n];
  v8f accs[4] = {acc0, acc1, acc2, acc3};
#pragma unroll
  for (int t = 0; t < 4; ++t) {
    float* op = out + (size_t)n * M + mBase + 16 * t + 8 * g;
#pragma unroll
    for (int i = 0; i < 8; ++i) op[i] = accs[t][i] + bb;
  }
}

// 2x2 max pool, stride 2 (single batch; out already offset to batch)
__global__ void pool2x2(const float* __restrict__ in, float* __restrict__ out,
                        int C, int Hc, int Wc, int Hp, int Wp) {
  int idx = blockIdx.x * blockDim.x + threadIdx.x;
  int tot = C * Hp * Wp;
  if (idx >= tot) return;
  int ow = idx % Wp;
  int t = idx / Wp;
  int oh = t % Hp;
  int c = t / Hp;
  const float* p = in + ((size_t)c * Hc + 2 * oh) * Wc + 2 * ow;
  out[idx] = fmaxf(fmaxf(p[0], p[1]), fmaxf(p[Wc], p[Wc + 1]));
}

// per-channel mean/var over (B, per) for x laid out (B, C, per)
__global__ void chan_stats(const float* __restrict__ x, float* __restrict__ stat,
                           int C, int per, int Bn) {
  int c = blockIdx.x;
  __shared__ float s1[256], s2[256];
  float a = 0.0f, b = 0.0f;
  int n = Bn * per;
  for (int i = threadIdx.x; i < n; i += blockDim.x) {
    int bb = i / per, j = i - bb * per;
    float v = x[((size_t)bb * C + c) * per + j];
    a += v; b += v * v;
  }
  s1[threadIdx.x] = a; s2[threadIdx.x] = b;
  __syncthreads();
  for (int st = blockDim.x / 2; st > 0; st >>= 1) {
    if ((int)threadIdx.x < st) { s1[threadIdx.x] += s1[threadIdx.x + st]; s2[threadIdx.x] += s2[threadIdx.x + st]; }
    __syncthreads();
  }
  if (threadIdx.x == 0) {
    float m = s1[0] / n;
    stat[c] = m;
    stat[C + c] = s2[0] / n - m * m;
  }
}

// per-channel mean/var over rows for x laid out (rows, D) (channel = last dim)
__global__ void chan_stats_lastdim(const float* __restrict__ x, float* __restrict__ stat,
                                   int D, int rows) {
  int d = blockIdx.x;
  __shared__ float s1[256], s2[256];
  float a = 0.0f, b = 0.0f;
  for (int i = threadIdx.x; i < rows; i += blockDim.x) {
    float v = x[(size_t)i * D + d];
    a += v; b += v * v;
  }
  s1[threadIdx.x] = a; s2[threadIdx.x] = b;
  __syncthreads();
  for (int st = blockDim.x / 2; st > 0; st >>= 1) {
    if ((int)threadIdx.x < st) { s1[threadIdx.x] += s1[threadIdx.x + st]; s2[threadIdx.x] += s2[threadIdx.x + st]; }
    __syncthreads();
  }
  if (threadIdx.x == 0) {
    float m = s1[0] / rows;
    stat[d] = m;
    stat[D + d] = s2[0] / rows - m * m;
  }
}

// x = (x - mean[c]) * rsqrt(var[c]+eps)   (x laid out (B,C,per))
__global__ void norm_apply(float* __restrict__ x, const float* __restrict__ stat,
                           int C, int per, int total) {
  int idx = blockIdx.x * blockDim.x + threadIdx.x;
  if (idx >= total) return;
  int c = (idx / per) % C;
  x[idx] = (x[idx] - stat[c]) * rsqrtf(stat[C + c] + EPSV);
}

// avg[b*C+c] = mean over per of |x|
__global__ void absmean_k(const float* __restrict__ x, float* __restrict__ avg, int per) {
  int bc = blockIdx.x;
  __shared__ float s1[256];
  float a = 0.0f;
  const float* p = x + (size_t)bc * per;
  for (int i = threadIdx.x; i < per; i += blockDim.x) a += fabsf(p[i]);
  s1[threadIdx.x] = a;
  __syncthreads();
  for (int st = blockDim.x / 2; st > 0; st >>= 1) {
    if ((int)threadIdx.x < st) s1[threadIdx.x] += s1[threadIdx.x + st];
    __syncthreads();
  }
  if (threadIdx.x == 0) avg[bc] = s1[0] / per;
}

// shrink FC: thr[b,c] = avg[b,c] * sigmoid(relu(BN_batch(avg @ fcw0^T + fcb0)) @ fcw1^T + fcb1)
__global__ void shrink_fc(const float* __restrict__ avg, const float* __restrict__ fcw,
                          const float* __restrict__ fcb, const float* __restrict__ bn,
                          float* __restrict__ thr, int Bn, int C) {
  __shared__ float h1[2048];
  __shared__ float h2[2048];
  __shared__ float mv[256], vv[256];
  const float* fcw0 = fcw;
  const float* fcw1 = fcw + (size_t)C * C;
  const float* fcb0 = fcb;
  const float* fcb1 = fcb + C;
  const float* g0 = bn;
  const float* b0 = bn + C;
  int tot = Bn * C;
  for (int idx = threadIdx.x; idx < tot; idx += blockDim.x) {
    int b = idx / C, i = idx - b * C;
    float s = fcb0[i];
    const float* wr = fcw0 + (size_t)i * C;
    const float* ar = avg + (size_t)b * C;
    for (int c = 0; c < C; ++c) s += ar[c] * wr[c];
    h1[idx] = s;
  }
  __syncthreads();
  for (int i = threadIdx.x; i < C; i += blockDim.x) {
    float s = 0.0f;
    for (int b = 0; b < Bn; ++b) s += h1[b * C + i];
    float m = s / Bn;
    float q = 0.0f;
    for (int b = 0; b < Bn; ++b) { float d = h1[b * C + i] - m; q += d * d; }
    mv[i] = m;
    vv[i] = q / Bn;
  }
  __syncthreads();
  for (int idx = threadIdx.x; idx < tot; idx += blockDim.x) {
    int i = idx % C;
    float hv = (h1[idx] - mv[i]) * rsqrtf(vv[i] + EPSV) * g0[i] + b0[i];
    h2[idx] = fmaxf(hv, 0.0f);
  }
  __syncthreads();
  for (int idx = threadIdx.x; idx < tot; idx += blockDim.x) {
    int b = idx / C, i = idx - b * C;
    float s = fcb1[i];
    const float* wr = fcw1 + (size_t)i * C;
    const float* hr = h2 + (size_t)b * C;
    for (int c = 0; c < C; ++c) s += hr[c] * wr[c];
    thr[idx] = avg[idx] * sigmoid_f(s);
  }
}

// soft threshold: x = sign(x) * max(|x| - thr[b,c], 0)
__global__ void shrink_apply(float* __restrict__ x, const float* __restrict__ thr,
                             int per, int total) {
  int idx = blockIdx.x * blockDim.x + threadIdx.x;
  if (idx >= total) return;
  float t = thr[idx / per];
  float v = x[idx];
  float a = fabsf(v) - t;
  x[idx] = (a > 0.0f) ? copysignf(a, v) : 0.0f;
}

// LayerNorm over last dim D; one block per row; safe for out == x
__global__ void layernorm_k(const float* __restrict__ x, const float* __restrict__ gamma,
                            const float* __restrict__ beta, float* __restrict__ out, int D) {
  int row = blockIdx.x;
  const float* xr = x + (size_t)row * D;
  __shared__ float s1[64], s2[64];
  __shared__ float mm, ii;
  float a = 0.0f, b = 0.0f;
  for (int d = threadIdx.x; d < D; d += blockDim.x) { float v = xr[d]; a += v; b += v * v; }
  s1[threadIdx.x] = a; s2[threadIdx.x] = b;
  __syncthreads();
  for (int st = blockDim.x / 2; st > 0; st >>= 1) {
    if ((int)threadIdx.x < st) { s1[threadIdx.x] += s1[threadIdx.x + st]; s2[threadIdx.x] += s2[threadIdx.x + st]; }
    __syncthreads();
  }
  if (threadIdx.x == 0) {
    float m = s1[0] / D;
    mm = m;
    ii = rsqrtf(s2[0] / D - m * m + EPSV);
  }
  __syncthreads();
  for (int d = threadIdx.x; d < D; d += blockDim.x)
    out[(size_t)row * D + d] = (xr[d] - mm) * ii * gamma[d] + beta[d];
}

// sliding-window attention (window +-2), q pre-scaled; x += out; probs -> d_out slice
__global__ void attn_k(const float* __restrict__ q, const float* __restrict__ k,
                       const float* __restrict__ v, float* __restrict__ x,
                       float* __restrict__ probs) {
  int idx = blockIdx.x * blockDim.x + threadIdx.x;
  int total = NBATCH * NHEADS * SEQ;
  if (idx >= total) return;
  int s = idx % SEQ;
  int t1 = idx / SEQ;
  int h = t1 % NHEADS;
  int b = t1 / NHEADS;
  const float* qp = q + ((size_t)b * SEQ + s) * DMODEL + h * HDIM;
  float sc[5];
#pragma unroll
  for (int w = 0; w < 5; ++w) {
    int t = s - 2 + w;
    if (t < 0 || t >= SEQ) { sc[w] = -1.0e9f; continue; }
    const float* kp = k + ((size_t)b * SEQ + t) * DMODEL + h * HDIM;
    float d = 0.0f;
    for (int j = 0; j < HDIM; ++j) d += qp[j] * kp[j];
    sc[w] = d;
  }
  float mx = sc[0];
#pragma unroll
  for (int w = 1; w < 5; ++w) mx = fmaxf(mx, sc[w]);
  float e[5], se = 0.0f;
#pragma unroll
  for (int w = 0; w < 5; ++w) { e[w] = __expf(sc[w] - mx); se += e[w]; }
  float inv = 1.0f / se;
  float p[5];
  float* pp = probs + (size_t)idx * 5;
#pragma unroll
  for (int w = 0; w < 5; ++w) { p[w] = e[w] * inv; pp[w] = p[w]; }
  float* xp = x + ((size_t)b * SEQ + s) * DMODEL + h * HDIM;
  for (int j = 0; j < HDIM; ++j) {
    float o = 0.0f;
#pragma unroll
    for (int w = 0; w < 5; ++w) {
      int t = s - 2 + w;
      t = t < 0 ? 0 : (t >= SEQ ? SEQ - 1 : t);
      o += p[w] * v[((size_t)b * SEQ + t) * DMODEL + h * HDIM + j];
    }
    xp[j] += o;
  }
}

// GLU: out[m,d] = y2[m,d] * sigmoid(y2[m, D+d]),  y2 is (M, 2D)
__global__ void glu_k(const float* __restrict__ y2, float* __restrict__ out, int D, int total) {
  int idx = blockIdx.x * blockDim.x + threadIdx.x;
  if (idx >= total) return;
  int m = idx / D, d = idx - m * D;
  float a = y2[(size_t)m * 2 * D + d];
  float g = y2[(size_t)m * 2 * D + D + d];
  out[idx] = a * sigmoid_f(g);
}

// depthwise conv along sequence (K taps, 'same' padding); layout (B*S, D)
__global__ void dwconv_k(const float* __restrict__ in, const float* __restrict__ w,
                         const float* __restrict__ bias, float* __restrict__ out,
                         int Ktap) {
  int idx = blockIdx.x * blockDim.x + threadIdx.x;
  int total = NTOK * DMODEL;
  if (idx >= total) return;
  int d = idx % DMODEL;
  int t = idx / DMODEL;
  int s = t % SEQ;
  int b = t / SEQ;
  int pad = (Ktap - 1) / 2;
  float acc = bias[d];
  const float* wr = w + (size_t)d * Ktap;
  for (int j = 0; j < Ktap; ++j) {
    int ss = s - pad + j;
    if (ss >= 0 && ss < SEQ) acc += in[((size_t)b * SEQ + ss) * DMODEL + d] * wr[j];
  }
  out[idx] = acc;
}

// batchnorm (per channel over rows) + swish; layout (rows, D)
__global__ void bnswish_k(const float* __restrict__ in, const float* __restrict__ stat,
                          const float* __restrict__ g, const float* __restrict__ b,
                          float* __restrict__ out, int D, int total) {
  int idx = blockIdx.x * blockDim.x + threadIdx.x;
  if (idx >= total) return;
  int d = idx % D;
  float v = (in[idx] - stat[d]) * rsqrtf(stat[D + d] + EPSV) * g[d] + b[d];
  out[idx] = v * sigmoid_f(v);
}

// repack stage-4 output (B, 256, 256, 3) NCHW -> (B*256 rows, 768) with k = c*3+w
__global__ void repack_k(const float* __restrict__ x4, float* __restrict__ out) {
  int idx = blockIdx.x * blockDim.x + threadIdx.x;
  int total = NBATCH * 256 * 768;
  if (idx >= total) return;
  int kk = idx % 768;
  int row = idx / 768;
  int hh = row % 256;
  int b = row / 256;
  int c = kk / 3;
  int w = kk - c * 3;
  out[idx] = x4[(((size_t)b * 256 + c) * 256 + hh) * 3 + w];
}

// ---------------------------------------------------------------------------
// workspace layout (floats)
// ---------------------------------------------------------------------------
static const size_t OFF_CONVBUF = 0;          // 7,864,320  (per-batch conv out, also proj input)
static const size_t OFF_POOLA   = 7864320;    // 15,728,640
static const size_t OFF_POOLB   = 23592960;   // 7,864,320
static const size_t OFF_LN      = 31457280;   // 294,912
static const size_t OFF_HID     = 31752192;   // 1,179,648
static const size_t OFF_Q       = 32931840;   // 294,912
static const size_t OFF_K       = 33226752;   // 294,912
static const size_t OFF_V       = 33521664;   // 294,912
static const size_t OFF_Y2      = 33816576;   // 589,824
static const size_t OFF_GLU     = 34406400;   // 294,912
static const size_t OFF_DW      = 34701312;   // 294,912
static const size_t OFF_STAT    = 34996224;   // 1,024
static const size_t OFF_ABS     = 34997248;   // 2,048
static const size_t OFF_THR     = 34999296;   // 2,048

extern "C" void kernel_launch(void* const* d_in, const int* in_sizes, int n_in,
                              void* d_out, int out_size, void* d_ws, size_t ws_size,
                              hipStream_t stream) {
  (void)in_sizes; (void)n_in; (void)out_size; (void)ws_size;
  const float* inputs = (const float*)d_in[0];
  float* ws = (float*)d_ws;

  float* convbuf = ws + OFF_CONVBUF;
  float* poolA = ws + OFF_POOLA;
  float* poolB = ws + OFF_POOLB;
  float* lnbuf = ws + OFF_LN;
  float* hid = ws + OFF_HID;
  float* qb = ws + OFF_Q;
  float* kb = ws + OFF_K;
  float* vb = ws + OFF_V;
  float* y2 = ws + OFF_Y2;
  float* glub = ws + OFF_GLU;
  float* dwb = ws + OFF_DW;
  float* statb = ws + OFF_STAT;
  float* absb = ws + OFF_ABS;
  float* thrb = ws + OFF_THR;

  // -------------------- front-end stages --------------------
  struct Stage {
    int Cin, Cout, Ks, pad, H, W;
    int iw_cw, iw_cb, iw_fcw, iw_fcb, iw_bn;
  };
  Stage st[4] = {
      {1, 32, 5, 2, 4096, 60, 1, 2, 3, 4, 5},
      {32, 64, 3, 1, 2048, 30, 6, 7, 8, 9, 10},
      {64, 128, 3, 1, 1024, 15, 11, 12, 13, 14, 15},
      {128, 256, 3, 1, 512, 7, 16, 17, 18, 19, 20},
  };
  const float* stage_in = inputs;
  float* pool_out[4] = {poolA, poolB, poolA, poolB};

  for (int s = 0; s < 4; ++s) {
    const Stage& P = st[s];
    const float* cw = (const float*)d_in[P.iw_cw];
    const float* cb = (const float*)d_in[P.iw_cb];
    const float* fcw = (const float*)d_in[P.iw_fcw];
    const float* fcb = (const float*)d_in[P.iw_fcb];
    const float* bn = (const float*)d_in[P.iw_bn];
    int Hc = P.H, Wc = P.W;
    int Hp = Hc / 2, Wp = Wc / 2;
    int Mconv = Hc * Wc;
    float* pout = pool_out[s];
    for (int b = 0; b < NBATCH; ++b) {
      const float* inb = stage_in + (size_t)b * P.Cin * Mconv;
      dim3 cgrid(Mconv / 64, P.Cout / 16);
      conv_wmma<<<cgrid, 32, 0, stream>>>(inb, cw, cb, convbuf, P.Cin, Hc, Wc, P.Cout, P.Ks, P.pad);
      int ptot = P.Cout * Hp * Wp;
      pool2x2<<<(ptot + 255) / 256, 256, 0, stream>>>(convbuf, pout + (size_t)b * ptot,
                                                      P.Cout, Hc, Wc, Hp, Wp);
    }
    int per = Hp * Wp;
    int total = NBATCH * P.Cout * per;
    chan_stats<<<P.Cout, 256, 0, stream>>>(pout, statb, P.Cout, per, NBATCH);
    norm_apply<<<(total + 255) / 256, 256, 0, stream>>>(pout, statb, P.Cout, per, total);
    absmean_k<<<NBATCH * P.Cout, 256, 0, stream>>>(pout, absb, per);
    shrink_fc<<<1, 256, 0, stream>>>(absb, fcw, fcb, bn, thrb, NBATCH, P.Cout);
    shrink_apply<<<(total + 255) / 256, 256, 0, stream>>>(pout, thrb, per, total);
    stage_in = pout;
  }

  // -------------------- projection --------------------
  float* X = (float*)d_out;                     // (2048, 144)
  float* attn_base = (float*)d_out + (size_t)NTOK * DMODEL;
  const float* proj_w = (const float*)d_in[21];
  const float* proj_b = (const float*)d_in[22];
  {
    int total = NBATCH * 256 * 768;
    repack_k<<<(total + 255) / 256, 256, 0, stream>>>(poolB, convbuf);
    dim3 g(NTOK / 64, DMODEL / 16);
    gemm_wmma<<<g, 32, 0, stream>>>(convbuf, proj_w, proj_b, X, nullptr,
                                    NTOK, DMODEL, 768, 1.0f, 0, 0.0f);
  }

  // -------------------- conformer layers --------------------
  const float* ln_params = (const float*)d_in[23];
  const float* ff_w1 = (const float*)d_in[24];
  const float* ff_b1 = (const float*)d_in[25];
  const float* ff_w2 = (const float*)d_in[26];
  const float* ff_b2 = (const float*)d_in[27];
  const float* qkv_w = (const float*)d_in[28];
  const float* qkv_b = (const float*)d_in[29];
  const float* pw1_w = (const float*)d_in[30];
  const float* pw1_b = (const float*)d_in[31];
  const float* dw_w = (const float*)d_in[32];
  const float* dw_b = (const float*)d_in[33];
  const float* pw2_w = (const float*)d_in[34];
  const float* pw2_b = (const float*)d_in[35];

  const int elems = NTOK * DMODEL;
  for (int l = 0; l < NLAYER; ++l) {
    const float* lnp = ln_params + (size_t)l * 5 * 2 * DMODEL;
    // --- FF half-step 1 ---
    layernorm_k<<<NTOK, 64, 0, stream>>>(X, lnp + 0 * 288, lnp + 0 * 288 + DMODEL, lnbuf, DMODEL);
    {
      dim3 g1(NTOK / 64, FFDIM / 16);
      gemm_wmma<<<g1, 32, 0, stream>>>(lnbuf, ff_w1 + (size_t)(l * 2 + 0) * FFDIM * DMODEL,
                                       ff_b1 + (size_t)(l * 2 + 0) * FFDIM, hid, nullptr,
                                       NTOK, FFDIM, DMODEL, 1.0f, 1, 0.0f);
      dim3 g2(NTOK / 64, DMODEL / 16);
      gemm_wmma<<<g2, 32, 0, stream>>>(hid, ff_w2 + (size_t)(l * 2 + 0) * DMODEL * FFDIM,
                                       ff_b2 + (size_t)(l * 2 + 0) * DMODEL, X, X,
                                       NTOK, DMODEL, FFDIM, 1.0f, 0, 0.5f);
    }
    // --- sliding attention ---
    {
      dim3 g(NTOK / 64, DMODEL / 16);
      gemm_wmma<<<g, 32, 0, stream>>>(X, qkv_w + (size_t)(l * 3 + 0) * DMODEL * DMODEL,
                                      qkv_b + (size_t)(l * 3 + 0) * DMODEL, qb, nullptr,
                                      NTOK, DMODEL, DMODEL, 1.0f / 6.0f, 0, 0.0f);
      gemm_wmma<<<g, 32, 0, stream>>>(X, qkv_w + (size_t)(l * 3 + 1) * DMODEL * DMODEL,
                                      qkv_b + (size_t)(l * 3 + 1) * DMODEL, kb, nullptr,
                                      NTOK, DMODEL, DMODEL, 1.0f, 0, 0.0f);
      gemm_wmma<<<g, 32, 0, stream>>>(X, qkv_w + (size_t)(l * 3 + 2) * DMODEL * DMODEL,
                                      qkv_b + (size_t)(l * 3 + 2) * DMODEL, vb, nullptr,
                                      NTOK, DMODEL, DMODEL, 1.0f, 0, 0.0f);
      int nthr = NBATCH * NHEADS * SEQ;
      attn_k<<<(nthr + 127) / 128, 128, 0, stream>>>(qb, kb, vb, X,
                                                     attn_base + (size_t)l * nthr * 5);
    }
    // --- conv module ---
    layernorm_k<<<NTOK, 64, 0, stream>>>(X, lnp + 1 * 288, lnp + 1 * 288 + DMODEL, lnbuf, DMODEL);
    {
      dim3 g1(NTOK / 64, (2 * DMODEL) / 16);
      gemm_wmma<<<g1, 32, 0, stream>>>(lnbuf, pw1_w + (size_t)l * 2 * DMODEL * DMODEL,
                                       pw1_b + (size_t)l * 2 * DMODEL, y2, nullptr,
                                       NTOK, 2 * DMODEL, DMODEL, 1.0f, 0, 0.0f);
      glu_k<<<(elems + 255) / 256, 256, 0, stream>>>(y2, glub, DMODEL, elems);
      dwconv_k<<<(elems + 255) / 256, 256, 0, stream>>>(glub, dw_w + (size_t)l * DMODEL * 31,
                                                        dw_b + (size_t)l * DMODEL, dwb, 31);
      chan_stats_lastdim<<<DMODEL, 256, 0, stream>>>(dwb, statb, DMODEL, NTOK);
      bnswish_k<<<(elems + 255) / 256, 256, 0, stream>>>(dwb, statb, lnp + 2 * 288,
                                                         lnp + 2 * 288 + DMODEL, glub,
                                                         DMODEL, elems);
      dim3 g2(NTOK / 64, DMODEL / 16);
      gemm_wmma<<<g2, 32, 0, stream>>>(glub, pw2_w + (size_t)l * DMODEL * DMODEL,
                                       pw2_b + (size_t)l * DMODEL, X, X,
                                       NTOK, DMODEL, DMODEL, 1.0f, 0, 1.0f);
    }
    // --- FF half-step 2 ---
    layernorm_k<<<NTOK, 64, 0, stream>>>(X, lnp + 3 * 288, lnp + 3 * 288 + DMODEL, lnbuf, DMODEL);
    {
      dim3 g1(NTOK / 64, FFDIM / 16);
      gemm_wmma<<<g1, 32, 0, stream>>>(lnbuf, ff_w1 + (size_t)(l * 2 + 1) * FFDIM * DMODEL,
                                       ff_b1 + (size_t)(l * 2 + 1) * FFDIM, hid, nullptr,
                                       NTOK, FFDIM, DMODEL, 1.0f, 1, 0.0f);
      dim3 g2(NTOK / 64, DMODEL / 16);
      gemm_wmma<<<g2, 32, 0, stream>>>(hid, ff_w2 + (size_t)(l * 2 + 1) * DMODEL * FFDIM,
                                       ff_b2 + (size_t)(l * 2 + 1) * DMODEL, X, X,
                                       NTOK, DMODEL, FFDIM, 1.0f, 0, 0.5f);
    }
    // --- final per-layer LN (in place) ---
    layernorm_k<<<NTOK, 64, 0, stream>>>(X, lnp + 4 * 288, lnp + 4 * 288 + DMODEL, X, DMODEL);
  }
}